// AttentionOnlyLlamaModel_27204322853635
// MI455X (gfx1250) — compile-verified
//
#include <hip/hip_runtime.h>
#include <hip/hip_bf16.h>
#include <math.h>

// ---------------------------------------------------------------------------
// Model constants (match reference)
// ---------------------------------------------------------------------------
static constexpr int LAYERS = 4;
static constexpr int D      = 1024;   // n_embd
static constexpr int H      = 16;     // heads
static constexpr int HD     = 64;     // head dim
static constexpr int B      = 2;
static constexpr int T      = 1024;
static constexpr int M      = B * T;  // rows into every projection GEMM
static constexpr float THETA = 10000.0f;
static constexpr float EPS   = 1e-5f;

typedef __attribute__((ext_vector_type(2))) float v2f;
typedef __attribute__((ext_vector_type(8))) float v8f;
typedef __attribute__((ext_vector_type(4))) int   v4i;

// ---------------------------------------------------------------------------
// CDNA5 async global->LDS staging (GLOBAL_LOAD_ASYNC_TO_LDS_B128, ASYNCcnt).
// Gated on builtin availability; falls back to synchronous b128 copies.
// Probe result (round 3): builtin exists, 4 args, arg0 is global int4*.
// ---------------------------------------------------------------------------
#if __has_builtin(__builtin_amdgcn_global_load_async_to_lds_b128) && \
    __has_builtin(__builtin_amdgcn_s_wait_asynccnt)
#define USE_ASYNC_LDS 1
#else
#define USE_ASYNC_LDS 0
#endif

__device__ __forceinline__ void stage16(const float* __restrict__ g,
                                        float* __restrict__ l) {
#if USE_ASYNC_LDS
  __builtin_amdgcn_global_load_async_to_lds_b128(
      (__attribute__((address_space(1))) v4i*)(g),
      (__attribute__((address_space(3))) v4i*)(l), /*offset=*/0, /*cpol=*/0);
#else
  *(float4*)l = *(const float4*)g;
#endif
}

__device__ __forceinline__ void stage_fence() {
#if USE_ASYNC_LDS
  __builtin_amdgcn_s_wait_asynccnt(0);   // wait for this wave's async copies
#endif
  __syncthreads();                       // all waves' copies now visible
}

// ---------------------------------------------------------------------------
// 1) Embedding gather: x[b,t,:] = embed[toks[b,t],:]
// ---------------------------------------------------------------------------
__global__ void gather_kernel(const int* __restrict__ toks,
                              const float* __restrict__ embed,
                              float* __restrict__ x) {
  size_t i = (size_t)blockIdx.x * blockDim.x + threadIdx.x;   // over M*D
  if (i >= (size_t)M * D) return;
  int row = (int)(i / D);
  int d   = (int)(i % D);
  int tok = toks[row];
  x[i] = embed[(size_t)tok * D + d];
}

// ---------------------------------------------------------------------------
// 2) LDS-tiled fp32 WMMA GEMM: Y[M,N] = X[M,K] @ W[K,N], row-major.
//    Block = 256 threads (8 waves).  Block tile: 128 (rows) x 64 (cols).
//    K-chunk = 16.  Each wave computes a 16x64 strip (4 v8f accumulators),
//    reusing its A-fragment across the 4 column tiles.
//
//    WMMA f32 16x16x4 ISA layouts (CDNA5 ISA §7.12.2):
//      A (16x4): lanes 0-15 rows 0..15 K=(0,1); lanes 16-31 same rows K=(2,3)
//      B (4x16): lanes 0-15 cols 0..15 K=(0,1); lanes 16-31 same cols K=(2,3)
//      C/D:      vgpr i -> row (i + 8*half), col = lane%16
// ---------------------------------------------------------------------------
__global__ void wmma_gemm_f32(const float* __restrict__ X,
                              const float* __restrict__ W,
                              float* __restrict__ Y,
                              int m, int n, int k) {
  __shared__ float Xs[128][17];   // +1 pad: odd stride -> conflict-free A reads
  __shared__ float Ws[16][65];    // +1 pad

  const int tiles_n = n >> 6;                 // blocks along N (64 cols each)
  const int tm = (blockIdx.x / tiles_n) << 7; // block row base (128)
  const int tn = (blockIdx.x % tiles_n) << 6; // block col base (64)

  const int tid  = threadIdx.x;
  const int wave = tid >> 5;                  // 0..7
  const int lane = tid & 31;
  const int half = lane >> 4;                 // K sub-offset selector
  const int l16  = lane & 15;
  const int wrow = wave << 4;                 // wave's row offset in block

  v8f acc[4] = {};

  for (int k0 = 0; k0 < k; k0 += 16) {
    __syncthreads();
    // Stage X panel: 128 rows x 16 cols, in 16-byte chunks (512 chunks).
    for (int idx = tid; idx < 512; idx += 256) {
      int r = idx >> 2, c = (idx & 3) << 2;
      stage16(X + (size_t)(tm + r) * k + k0 + c, &Xs[r][c]);
    }
    // Stage W panel: 16 rows x 64 cols (256 chunks).
    {
      int r = tid >> 4, c = (tid & 15) << 2;
      stage16(W + (size_t)(k0 + r) * n + tn + c, &Ws[r][c]);
    }
    // Prefetch next panels into cache while we compute this one.
    if (k0 + 16 < k) {
      __builtin_prefetch(X + (size_t)(tm + (tid >> 1)) * k + k0 + 16, 0, 0);
      __builtin_prefetch(W + (size_t)(k0 + 16 + (tid >> 4)) * n + tn, 0, 0);
    }
    stage_fence();

#pragma unroll
    for (int ks = 0; ks < 16; ks += 4) {
      const int ka = ks + 2 * half;
      v2f a = { Xs[wrow + l16][ka], Xs[wrow + l16][ka + 1] };
#pragma unroll
      for (int ct = 0; ct < 4; ++ct) {
        v2f b = { Ws[ka][(ct << 4) + l16], Ws[ka + 1][(ct << 4) + l16] };
        acc[ct] = __builtin_amdgcn_wmma_f32_16x16x4_f32(
            false, a, false, b, (short)0, acc[ct], false, false);
      }
    }
  }

#pragma unroll
  for (int ct = 0; ct < 4; ++ct) {
#pragma unroll
    for (int i = 0; i < 8; ++i) {
      int row = tm + wrow + i + (half << 3);
      Y[(size_t)row * n + tn + (ct << 4) + l16] = acc[ct][i];
    }
  }
}

// ---------------------------------------------------------------------------
// 3) RoPE + head transpose.
//    q/k/v_lin : [B, T, H*HD]  ->  q/k/v_r : [B, H, T, HD]
// ---------------------------------------------------------------------------
__global__ void rope_transpose_kernel(const float* __restrict__ qlin,
                                      const float* __restrict__ klin,
                                      const float* __restrict__ vlin,
                                      float* __restrict__ qr,
                                      float* __restrict__ kr,
                                      float* __restrict__ vr) {
  size_t i = (size_t)blockIdx.x * blockDim.x + threadIdx.x; // over B*T*H*(HD/2)
  constexpr size_t TOTAL = (size_t)B * T * H * (HD / 2);
  if (i >= TOTAL) return;

  int p = (int)(i % (HD / 2));
  int h = (int)((i / (HD / 2)) % H);
  int t = (int)((i / ((HD / 2) * H)) % T);
  int b = (int)(i / ((size_t)(HD / 2) * H * T));

  float inv = powf(THETA, -(2.0f * p) / (float)HD);
  float ang = (float)t * inv;
  float c = cosf(ang), s = sinf(ang);

  size_t src = (((size_t)b * T + t) * H + h) * HD + 2 * p;   // [B,T,H,HD]
  size_t dst = (((size_t)b * H + h) * T + t) * HD + 2 * p;   // [B,H,T,HD]

  float q0 = qlin[src], q1 = qlin[src + 1];
  qr[dst]     = q0 * c - q1 * s;
  qr[dst + 1] = q0 * s + q1 * c;

  float k0 = klin[src], k1 = klin[src + 1];
  kr[dst]     = k0 * c - k1 * s;
  kr[dst + 1] = k0 * s + k1 * c;

  vr[dst]     = vlin[src];
  vr[dst + 1] = vlin[src + 1];
}

// ---------------------------------------------------------------------------
// 4) WMMA flash attention (fp32, causal, online softmax).
//    Block = 128 threads (4 waves) owns 64 query rows of one (b,h).
//    Per 16-key tile:
//      S  = Q(16x64) @ K^T(64x16)   -> 16x v_wmma_f32_16x16x4_f32
//      softmax update via half-wave __shfl_xor row reductions
//      O += P(16x16) @ V(16x64)     -> 16x WMMA (P bounced via LDS to
//                                      convert D-layout -> A-layout)
//    Output written directly in [B,T,H*HD] layout (fused inverse transpose).
// ---------------------------------------------------------------------------
__global__ void flash_attn_kernel(const float* __restrict__ qr,
                                  const float* __restrict__ kr,
                                  const float* __restrict__ vr,
                                  float* __restrict__ olin) {
  constexpr int QB = 64;                  // query rows per block
  __shared__ float Ks[16][65];            // [key][hd]
  __shared__ float Vs[16][65];            // [key][hd]
  __shared__ float Ps[4][16][17];         // per-wave P tile [row][key]

  const int qblocks = T / QB;
  const int bh    = blockIdx.x / qblocks;        // b*H + h
  const int qblk  = blockIdx.x % qblocks;
  const int qbase = qblk * QB;
  const int b = bh / H;
  const int h = bh % H;

  const int tid  = threadIdx.x;
  const int wave = tid >> 5;              // 0..3
  const int lane = tid & 31;
  const int half = lane >> 4;
  const int l16  = lane & 15;
  const int qw   = qbase + (wave << 4);   // this wave's 16 query rows

  const float* Qh = qr + (size_t)bh * T * HD;
  const float* Kh = kr + (size_t)bh * T * HD;
  const float* Vh = vr + (size_t)bh * T * HD;

  // Preload Q A-fragments for all 16 K-steps of the HD=64 reduction.
  v2f qa[16];
#pragma unroll
  for (int s = 0; s < 16; ++s) {
    const float* qp = Qh + (size_t)(qw + l16) * HD + 4 * s + 2 * half;
    qa[s] = { qp[0], qp[1] };
  }

  v8f oacc[4] = {};
  float m_i[8], l_i[8];
#pragma unroll
  for (int i = 0; i < 8; ++i) { m_i[i] = -INFINITY; l_i[i] = 0.0f; }

  const int ntiles = (qbase + QB) / 16;   // keys 0 .. qbase+63

  for (int kt = 0; kt < ntiles; ++kt) {
    const int kb = kt * 16;
    __syncthreads();
    // Stage K/V tiles (16 x 64 each; 256 16-byte chunks per array).
    for (int idx = tid; idx < 256; idx += 128) {
      int r = idx >> 4, c = (idx & 15) << 2;
      stage16(Kh + (size_t)(kb + r) * HD + c, &Ks[r][c]);
      stage16(Vh + (size_t)(kb + r) * HD + c, &Vs[r][c]);
    }
    stage_fence();

    // ---- S = Q @ K^T  (B fragment: B[kdim][key] = K[key][kdim]) ----
    v8f s_acc = {};
#pragma unroll
    for (int s = 0; s < 16; ++s) {
      const int kk = 4 * s + 2 * half;
      v2f bfr = { Ks[l16][kk], Ks[l16][kk + 1] };
      s_acc = __builtin_amdgcn_wmma_f32_16x16x4_f32(
          false, qa[s], false, bfr, (short)0, s_acc, false, false);
    }

    // ---- scale, causal mask, online softmax ----
    float corr[8];
    const int j = kb + l16;               // this lane's key column
#pragma unroll
    for (int i = 0; i < 8; ++i) {
      const int r = qw + i + (half << 3); // this lane's query row for vgpr i
      float s = s_acc[i] * 0.125f;        // 1/sqrt(64)
      if (j > r) s = -INFINITY;
      // row max over the 16 lanes of this half-wave
      float mx = s;
#pragma unroll
      for (int off = 1; off <= 8; off <<= 1)
        mx = fmaxf(mx, __shfl_xor(mx, off, 32));
      const float mnew = fmaxf(m_i[i], mx);
      const float p = expf(s - mnew);     // exp(-inf) = 0 on masked lanes
      corr[i] = expf(m_i[i] - mnew);      // first tile: exp(-inf) = 0
      float ps = p;
#pragma unroll
      for (int off = 1; off <= 8; off <<= 1)
        ps += __shfl_xor(ps, off, 32);
      l_i[i] = l_i[i] * corr[i] + ps;
      m_i[i] = mnew;
      Ps[wave][i + (half << 3)][l16] = p; // D-layout -> LDS
    }

    // rescale O accumulators by the softmax correction
#pragma unroll
    for (int ct = 0; ct < 4; ++ct)
#pragma unroll
      for (int i = 0; i < 8; ++i)
        oacc[ct][i] *= corr[i];

    // ---- O += P @ V  (A: P[row][key] from LDS; B: V[key][hd]) ----
#pragma unroll
    for (int s2 = 0; s2 < 4; ++s2) {
      const int kk = 4 * s2 + 2 * half;
      v2f a = { Ps[wave][l16][kk], Ps[wave][l16][kk + 1] };
#pragma unroll
      for (int ct = 0; ct < 4; ++ct) {
        v2f bfr = { Vs[kk][(ct << 4) + l16], Vs[kk + 1][(ct << 4) + l16] };
        oacc[ct] = __builtin_amdgcn_wmma_f32_16x16x4_f32(
            false, a, false, bfr, (short)0, oacc[ct], false, false);
      }
    }
  }

  // ---- normalize and write out in [B,T,H*HD] layout ----
#pragma unroll
  for (int i = 0; i < 8; ++i) {
    const int row = qw + i + (half << 3);
    const float invs = 1.0f / l_i[i];
    float* orow = olin + (((size_t)b * T + row) * H + h) * HD;
#pragma unroll
    for (int ct = 0; ct < 4; ++ct)
      orow[(ct << 4) + l16] = oacc[ct][i] * invs;
  }
}

// ---------------------------------------------------------------------------
// 5) Final RMSNorm: one 256-thread block per (b,t) row.
// ---------------------------------------------------------------------------
__global__ void rmsnorm_kernel(const float* __restrict__ x,
                               const float* __restrict__ w,
                               float* __restrict__ out) {
  __shared__ float red[256];
  int row = blockIdx.x;
  const float* xr = x + (size_t)row * D;

  float ss = 0.0f;
  for (int d = threadIdx.x; d < D; d += blockDim.x) {
    float v = xr[d];
    ss += v * v;
  }
  red[threadIdx.x] = ss;
  __syncthreads();
  for (int s = 128; s > 0; s >>= 1) {
    if (threadIdx.x < (unsigned)s) red[threadIdx.x] += red[threadIdx.x + s];
    __syncthreads();
  }
  float inv = rsqrtf(red[0] / (float)D + EPS);
  for (int d = threadIdx.x; d < D; d += blockDim.x) {
    out[(size_t)row * D + d] = w[d] * xr[d] * inv;
  }
}

// ---------------------------------------------------------------------------
// Host-side orchestration (graph-capture safe: kernel launches only)
// ---------------------------------------------------------------------------
extern "C" void kernel_launch(void* const* d_in, const int* in_sizes, int n_in,
                              void* d_out, int out_size, void* d_ws, size_t ws_size,
                              hipStream_t stream) {
  const int*   toks   = (const int*)  d_in[0];
  const float* embed  = (const float*)d_in[1];
  const float* Wq     = (const float*)d_in[2];
  const float* Wk     = (const float*)d_in[3];
  const float* Wv     = (const float*)d_in[4];
  const float* Wo     = (const float*)d_in[5];
  const float* norm_w = (const float*)d_in[6];
  float*       out    = (float*)d_out;

  // Workspace carve-up: 7 buffers of M*D floats (8 MB each).
  const size_t S = (size_t)M * D;
  float* ws   = (float*)d_ws;
  float* x    = ws + 0 * S;
  float* qlin = ws + 1 * S;   // reused as attention output (olin)
  float* klin = ws + 2 * S;
  float* vlin = ws + 3 * S;
  float* qrb  = ws + 4 * S;
  float* krb  = ws + 5 * S;
  float* vrb  = ws + 6 * S;
  float* olin = qlin;

  // ---- embedding gather ----
  {
    int threads = 256;
    int blocks  = (int)(((size_t)M * D + threads - 1) / threads);
    gather_kernel<<<blocks, threads, 0, stream>>>(toks, embed, x);
  }

  const int gemm_blocks = (M / 128) * (D / 64);        // 256 blocks
  const int attn_blocks = B * H * (T / 64);            // 512 blocks

  for (int l = 0; l < LAYERS; ++l) {
    const float* wq = Wq + (size_t)l * D * D;
    const float* wk = Wk + (size_t)l * D * D;
    const float* wv = Wv + (size_t)l * D * D;
    const float* wo = Wo + (size_t)l * D * D;

    // Q/K/V projections (LDS-tiled WMMA)
    wmma_gemm_f32<<<gemm_blocks, 256, 0, stream>>>(x, wq, qlin, M, D, D);
    wmma_gemm_f32<<<gemm_blocks, 256, 0, stream>>>(x, wk, klin, M, D, D);
    wmma_gemm_f32<<<gemm_blocks, 256, 0, stream>>>(x, wv, vlin, M, D, D);

    // RoPE + transpose to [B,H,T,HD]
    {
      size_t total = (size_t)B * T * H * (HD / 2);
      int threads = 256;
      int blocks  = (int)((total + threads - 1) / threads);
      rope_transpose_kernel<<<blocks, threads, 0, stream>>>(qlin, klin, vlin,
                                                            qrb, krb, vrb);
    }

    // WMMA flash attention (writes olin in [B,T,D] layout)
    flash_attn_kernel<<<attn_blocks, 128, 0, stream>>>(qrb, krb, vrb, olin);

    // Output projection: x <- olin @ Wo  (no residual, per reference)
    wmma_gemm_f32<<<gemm_blocks, 256, 0, stream>>>(olin, wo, x, M, D, D);
  }

  // Final RMSNorm
  rmsnorm_kernel<<<M, 256, 0, stream>>>(x, norm_w, out);
}